// fisher_model_55765855372194
// MI455X (gfx1250) — compile-verified
//
#include <hip/hip_runtime.h>
#include <math.h>

// ---------------------------------------------------------------------------
// MPNN energy + analytic force kernel for MI455X (gfx1250, wave32).
// One workgroup (256 threads = 8 waves) per molecule (64 atoms, 4032 edges).
// - node features x[64][4][32] live in LDS (320KB/WGP on CDNA5)
// - edge phase: 1 wave per edge, lane = feature (FEAT==32==wave32)
// - node MLP (32x32) via V_WMMA_F32_16X16X4_F32, K chained 8x (K=32)
// - segment sums via ds_add_f32 LDS atomics
// - forward activations spilled to d_ws (~11MB total: L2-resident, 192MB L2)
// - backward derived by hand; only position grads are needed
// ---------------------------------------------------------------------------

typedef float v2f __attribute__((ext_vector_type(2)));
typedef float v8f __attribute__((ext_vector_type(8)));

#define FS 33            // padded feature row stride (kills LDS bank conflicts)
#define AS 132           // atom stride = 4*FS
#define NATOM 64         // atoms per molecule (fixed by generator)
#define BLK 256          // 8 waves

__device__ __forceinline__ float sigm(float x) { return 1.f / (1.f + __expf(-x)); }

__device__ __forceinline__ float wave_sum(float v) {
  #pragma unroll
  for (int off = 16; off > 0; off >>= 1) v += __shfl_xor(v, off, 32);
  return v;
}

// Bernstein radial basis (gammaln terms == log C(15,k), compile-time consts)
// rad[b] = C(15,b) u^b (1-u)^(15-b)   (cutoff applied by caller)
template <bool BWD>
__device__ __forceinline__ void edge_basis(float dx, float dy, float dz,
    float& r, float& ir, float& cut, float& Y1, float& Y2, float& Y3,
    float rad[16], float dradb[16], float& cutp, float& dudr)
{
  const float BC[16] = {1.f, 15.f, 105.f, 455.f, 1365.f, 3003.f, 5005.f, 6435.f,
                        6435.f, 5005.f, 3003.f, 1365.f, 455.f, 105.f, 15.f, 1.f};
  float r2 = dx * dx + dy * dy + dz * dz + 1e-12f;
  r = sqrtf(r2);
  ir = 1.f / r;
  float uraw = 1.f / (1.f + r);
  float u = fminf(fmaxf(uraw, 1e-7f), 1.f - 1e-7f);
  float v = 1.f - u;
  float x2r = r2 * 0.04f;                       // (r/5)^2
  float x2 = fminf(x2r, 1.f - 1e-6f);
  bool inCut = (r < 5.0f);
  float om = 1.f - x2;
  cut = inCut ? __expf(1.f - 1.f / om) : 0.f;
  Y1 = dx * ir; Y2 = dy * ir; Y3 = dz * ir;
  float up[16], vp[16];
  up[0] = 1.f; vp[0] = 1.f;
  #pragma unroll
  for (int b = 1; b < 16; ++b) { up[b] = up[b - 1] * u; vp[b] = vp[b - 1] * v; }
  #pragma unroll
  for (int b = 0; b < 16; ++b) rad[b] = BC[b] * up[b] * vp[15 - b];
  if (BWD) {
    #pragma unroll
    for (int b = 0; b < 16; ++b) {
      float t = 0.f;
      if (b > 0)  t += (float)b * up[b - 1] * vp[15 - b];
      if (b < 15) t -= (float)(15 - b) * up[b] * vp[14 - b];
      dradb[b] = BC[b] * t;                     // d rad / du
    }
    dudr = (uraw > 1e-7f && uraw < 1.f - 1e-7f) ? -u * u : 0.f;
    cutp = (inCut && x2r < 1.f - 1e-6f) ? -cut * (2.f * r * 0.04f) / (om * om) : 0.f;
  }
}

// OUT[row][col] = sum_k IN[row][k] * W(k,col), 32x32 weights (padded FS),
// rows RM (multiple of 16), via V_WMMA_F32_16X16X4_F32, K chained 8x.
// biasEvery: 0=no bias, 4=rows with l==0 (flattened stride 4), 1=all rows.
__device__ __forceinline__ void mm32(const float* __restrict__ in, int inStride,
                                     float* __restrict__ out, int outStride,
                                     const float* __restrict__ w, bool transW,
                                     int RM, const float* __restrict__ bias, int biasEvery)
{
  const int tid = threadIdx.x;
  const int wave = tid >> 5, lane = tid & 31;
  const int l15 = lane & 15, lh = lane >> 4;
  const int tiles = (RM >> 4) * 2;
  for (int t = wave; t < tiles; t += 8) {
    const int mt = t >> 1;
    const int nt = (t & 1) << 4;
    const int arow = (mt << 4) + l15;
    const int bcol = nt + l15;
    v8f acc = {0.f, 0.f, 0.f, 0.f, 0.f, 0.f, 0.f, 0.f};
    #pragma unroll
    for (int kk = 0; kk < 8; ++kk) {
      const int ka = (kk << 2) + (lh << 1);     // lanes 0-15: K=4kk+{0,1}; 16-31: +{2,3}
      v2f a, b;
      a.x = in[arow * inStride + ka];
      a.y = in[arow * inStride + ka + 1];
      if (transW) { b.x = w[bcol * FS + ka];  b.y = w[bcol * FS + ka + 1]; }
      else        { b.x = w[ka * FS + bcol];  b.y = w[(ka + 1) * FS + bcol]; }
      acc = __builtin_amdgcn_wmma_f32_16x16x4_f32(false, a, false, b, (short)0, acc,
                                                  false, false);
    }
    #pragma unroll
    for (int j = 0; j < 8; ++j) {               // VGPR j: M=j (lanes 0-15) / M=j+8
      const int row = (mt << 4) + (lh << 3) + j;
      float vv = acc[j];
      if (biasEvery > 0 && (row % biasEvery) == 0) vv += bias[bcol];
      out[row * outStride + bcol] = vv;
    }
  }
}

// -------- forward edge phase: accumulate messages into sY via ds atomics ----
template <bool LAST>
__device__ __forceinline__ void edge_fwd(int epm, const int* __restrict__ dstI,
    const int* __restrict__ srcI, int edgeBase, int atomBase,
    const float* __restrict__ sPos, const float* __restrict__ sX,
    float* __restrict__ sY, const float* __restrict__ sWr, const float* __restrict__ sWb)
{
  const int wave = threadIdx.x >> 5, lane = threadIdx.x & 31;
  for (int e = wave; e < epm; e += 8) {
    const int d = dstI[edgeBase + e] - atomBase;
    const int s = srcI[edgeBase + e] - atomBase;
    float dx = sPos[s * 3 + 0] - sPos[d * 3 + 0];
    float dy = sPos[s * 3 + 1] - sPos[d * 3 + 1];
    float dz = sPos[s * 3 + 2] - sPos[d * 3 + 2];
    float r, ir, cut, Y1, Y2, Y3, cutp, dudr;
    float rad[16], dd[16];
    edge_basis<false>(dx, dy, dz, r, ir, cut, Y1, Y2, Y3, rad, dd, cutp, dudr);
    float Rraw = 0.f, Rbraw = 0.f;
    #pragma unroll
    for (int b = 0; b < 16; ++b) {
      Rraw  += rad[b] * sWr[b * FS + lane];
      Rbraw += rad[b] * sWb[b * FS + lane];
    }
    float R = Rraw * cut, Rb = Rbraw * cut;
    float xs0 = sX[s * AS + lane];
    if (!LAST) {
      float xs1 = sX[s * AS + FS + lane];
      float xs2 = sX[s * AS + 2 * FS + lane];
      float xs3 = sX[s * AS + 3 * FS + lane];
      float by = Rb * xs0;
      atomicAdd(&sY[d * AS + lane],          R * xs0 + by);      // Y0 == 1
      atomicAdd(&sY[d * AS + FS + lane],     R * xs1 + Y1 * by);
      atomicAdd(&sY[d * AS + 2 * FS + lane], R * xs2 + Y2 * by);
      atomicAdd(&sY[d * AS + 3 * FS + lane], R * xs3 + Y3 * by);
    } else {
      float S = xs0 + Y1 * sX[s * AS + FS + lane]
                    + Y2 * sX[s * AS + 2 * FS + lane]
                    + Y3 * sX[s * AS + 3 * FS + lane];
      atomicAdd(&sY[d * AS + lane], R * xs0 + Rb * S);
    }
  }
}

// -------- backward edge phase: grads to source features and positions -------
template <bool LAST>
__device__ __forceinline__ void edge_bwd(int epm, const int* __restrict__ dstI,
    const int* __restrict__ srcI, int edgeBase, int atomBase,
    const float* __restrict__ sPos, const float* __restrict__ gy,
    float* __restrict__ gx, float* __restrict__ sGp,
    const float* __restrict__ xws,            // saved x_i, global (L2-resident)
    const float* __restrict__ sWr, const float* __restrict__ sWb)
{
  const int wave = threadIdx.x >> 5, lane = threadIdx.x & 31;
  for (int e = wave; e < epm; e += 8) {
    const int d = dstI[edgeBase + e] - atomBase;
    const int s = srcI[edgeBase + e] - atomBase;
    float dx = sPos[s * 3 + 0] - sPos[d * 3 + 0];
    float dy = sPos[s * 3 + 1] - sPos[d * 3 + 1];
    float dz = sPos[s * 3 + 2] - sPos[d * 3 + 2];
    float r, ir, cut, Y1, Y2, Y3, cutp, dudr;
    float rad[16], dradb[16];
    edge_basis<true>(dx, dy, dz, r, ir, cut, Y1, Y2, Y3, rad, dradb, cutp, dudr);
    float Rraw = 0.f, Rbraw = 0.f, dRw = 0.f, dRbw = 0.f;
    #pragma unroll
    for (int b = 0; b < 16; ++b) {
      float wrv = sWr[b * FS + lane], wbv = sWb[b * FS + lane];
      Rraw  += rad[b] * wrv;   Rbraw += rad[b] * wbv;
      dRw   += dradb[b] * wrv; dRbw  += dradb[b] * wbv;
    }
    float R = Rraw * cut, Rb = Rbraw * cut;
    const float* xsrow = xws + (size_t)(atomBase + s) * 128;
    float xs0 = xsrow[lane], xs1 = xsrow[32 + lane];
    float xs2 = xsrow[64 + lane], xs3 = xsrow[96 + lane];
    float gR, gRb, p1, p2, p3;
    if (!LAST) {
      float gm0 = gy[d * AS + lane],          gm1 = gy[d * AS + FS + lane];
      float gm2 = gy[d * AS + 2 * FS + lane], gm3 = gy[d * AS + 3 * FS + lane];
      float sYg = gm0 + gm1 * Y1 + gm2 * Y2 + gm3 * Y3;
      atomicAdd(&gx[s * AS + lane],          gm0 * R + sYg * Rb);
      atomicAdd(&gx[s * AS + FS + lane],     gm1 * R);
      atomicAdd(&gx[s * AS + 2 * FS + lane], gm2 * R);
      atomicAdd(&gx[s * AS + 3 * FS + lane], gm3 * R);
      gR  = gm0 * xs0 + gm1 * xs1 + gm2 * xs2 + gm3 * xs3;
      gRb = sYg * xs0;
      p1 = gm1 * Rb * xs0; p2 = gm2 * Rb * xs0; p3 = gm3 * Rb * xs0;
    } else {
      float gm = gy[d * AS + lane];
      float S = xs0 + Y1 * xs1 + Y2 * xs2 + Y3 * xs3;
      float gmRb = gm * Rb;
      atomicAdd(&gx[s * AS + lane],          gm * R + gmRb);     // Y0 == 1
      atomicAdd(&gx[s * AS + FS + lane],     gmRb * Y1);
      atomicAdd(&gx[s * AS + 2 * FS + lane], gmRb * Y2);
      atomicAdd(&gx[s * AS + 3 * FS + lane], gmRb * Y3);
      gR = gm * xs0; gRb = gm * S;
      p1 = gmRb * xs1; p2 = gmRb * xs2; p3 = gmRb * xs3;
    }
    // collapse the per-edge scalar grads to 5 wave reductions:
    //   g_cut = sum_f (gR*Rraw + gRb*Rbraw);  g_u = cut * sum_f (gR*dRw + gRb*dRbw)
    float s1 = wave_sum(gR * Rraw + gRb * Rbraw);
    float s2 = wave_sum(gR * dRw + gRb * dRbw);
    float gY1 = wave_sum(p1), gY2 = wave_sum(p2), gY3 = wave_sum(p3);
    float g_r = (s2 * cut) * dudr + s1 * cutp;
    float k3 = (gY1 * dx + gY2 * dy + gY3 * dz) * ir * ir * ir;
    float gDx = gY1 * ir - dx * k3 + g_r * dx * ir;
    float gDy = gY2 * ir - dy * k3 + g_r * dy * ir;
    float gDz = gY3 * ir - dz * k3 + g_r * dz * ir;
    float gD = (lane == 0) ? gDx : (lane == 1) ? gDy : gDz;
    if (lane < 3) {
      atomicAdd(&sGp[s * 3 + lane], gD);
      atomicAdd(&sGp[d * 3 + lane], -gD);
    }
  }
}

__device__ __forceinline__ void loadW(int i,
    const float* __restrict__ wr, const float* __restrict__ wb,
    const float* __restrict__ w1, const float* __restrict__ b1,
    const float* __restrict__ w2, const float* __restrict__ b2,
    float* sWr, float* sWb, float* sW1, float* sW2, float* sB1, float* sB2)
{
  const int tid = threadIdx.x;
  for (int idx = tid; idx < 512; idx += BLK) {
    int b = idx >> 5, f = idx & 31;
    sWr[b * FS + f] = wr[i * 512 + idx];
    sWb[b * FS + f] = wb[i * 512 + idx];
  }
  for (int idx = tid; idx < 1024; idx += BLK) {
    int k = idx >> 5, c = idx & 31;
    sW1[k * FS + c] = w1[i * 1024 + idx];
    sW2[k * FS + c] = w2[i * 1024 + idx];
  }
  if (tid < 32) { sB1[tid] = b1[i * 32 + tid]; sB2[tid] = b2[i * 32 + tid]; }
  if (tid == 0 && i < 2) {                    // warm GL2 for next iter's weights
    __builtin_prefetch(wr + (i + 1) * 512, 0, 1);
    __builtin_prefetch(w1 + (i + 1) * 1024, 0, 1);
    __builtin_prefetch(w2 + (i + 1) * 1024, 0, 1);
  }
}

__global__ __launch_bounds__(BLK)
void fisher_mpnn_kernel(const int* __restrict__ Z, const float* __restrict__ pos,
    const int* __restrict__ dstI, const int* __restrict__ srcI,
    const float* __restrict__ embed, const float* __restrict__ wr,
    const float* __restrict__ wb, const float* __restrict__ w1,
    const float* __restrict__ b1, const float* __restrict__ w2,
    const float* __restrict__ b2, const float* __restrict__ w_out,
    const float* __restrict__ ebias, float* __restrict__ out,
    float* __restrict__ ws, int N, int NA, int EPM, int BS)
{
  __shared__ float sX[NATOM * AS];            // x / gx        (padded)
  __shared__ float sY[NATOM * AS];            // y / t2 / g_t1
  __shared__ float sT[NATOM * AS];            // t1 / t3 / g_y
  __shared__ float sWr[16 * FS], sWb[16 * FS];
  __shared__ float sW1[32 * FS], sW2[32 * FS];
  __shared__ float sB1[32], sB2[32], sWout[32], gt2v[32];
  __shared__ float sPos[NATOM * 3], sGp[NATOM * 3];
  __shared__ float sEnergy;

  const int tid = threadIdx.x, wave = tid >> 5, lane = tid & 31;
  const int mol = blockIdx.x;
  const int atomBase = mol * NA, edgeBase = mol * EPM;

  float* xws  = ws;                               // [3][N][4][32]
  float* t1ws = xws + (size_t)3 * N * 128;        // [2][N][4][32]
  float* t12w = t1ws + (size_t)2 * N * 128;       // [N][32]

  for (int idx = tid; idx < NATOM * 3; idx += BLK) {
    sPos[idx] = pos[atomBase * 3 + idx];
    sGp[idx] = 0.f;
  }
  if (tid < 32) sWout[tid] = w_out[tid];
  if (tid == 0) sEnergy = 0.f;
  for (int idx = tid; idx < NATOM * 128; idx += BLK) {
    int a = idx >> 7, rem = idx & 127, l = rem >> 5, f = rem & 31;
    sX[a * AS + l * FS + f] = (l == 0) ? embed[Z[atomBase + a] * 32 + f] : 0.f;
  }
  __syncthreads();

  // ======================= forward, iterations 0 and 1 =======================
  for (int it = 0; it < 2; ++it) {
    loadW(it, wr, wb, w1, b1, w2, b2, sWr, sWb, sW1, sW2, sB1, sB2);
    for (int idx = tid; idx < NATOM * 128; idx += BLK) {
      int a = idx >> 7, rem = idx & 127, l = rem >> 5, f = rem & 31;
      float xv = sX[a * AS + l * FS + f];
      xws[(size_t)it * N * 128 + (size_t)(atomBase + a) * 128 + l * 32 + f] = xv;
      sY[a * AS + l * FS + f] = xv;               // y = x + segsum
    }
    __syncthreads();
    edge_fwd<false>(EPM, dstI, srcI, edgeBase, atomBase, sPos, sX, sY, sWr, sWb);
    __syncthreads();
    mm32(sY, FS, sT, FS, sW1, false, 256, sB1, 4);        // t1 = y@w1 (+b1 @ l=0)
    __syncthreads();
    for (int idx = tid; idx < NATOM * 128; idx += BLK) {  // save t1; t2 = t1*sig(t1_0)
      int a = idx >> 7, rem = idx & 127, l = rem >> 5, f = rem & 31;
      float t1v = sT[a * AS + l * FS + f];
      t1ws[(size_t)it * N * 128 + (size_t)(atomBase + a) * 128 + l * 32 + f] = t1v;
      float g = sigm(sT[a * AS + f]);
      sY[a * AS + l * FS + f] = t1v * g;
    }
    __syncthreads();
    mm32(sY, FS, sT, FS, sW2, false, 256, sB2, 4);        // t3 = t2@w2 (+b2 @ l=0)
    __syncthreads();
    for (int idx = tid; idx < NATOM * 128; idx += BLK) {  // x += t3
      int a = idx >> 7, rem = idx & 127, l = rem >> 5, f = rem & 31;
      sX[a * AS + l * FS + f] += sT[a * AS + l * FS + f];
    }
    __syncthreads();
  }

  // ============================ forward, iteration 2 =========================
  loadW(2, wr, wb, w1, b1, w2, b2, sWr, sWb, sW1, sW2, sB1, sB2);
  for (int idx = tid; idx < NATOM * 128; idx += BLK) {
    int a = idx >> 7, rem = idx & 127, l = rem >> 5, f = rem & 31;
    xws[(size_t)2 * N * 128 + (size_t)(atomBase + a) * 128 + l * 32 + f] =
        sX[a * AS + l * FS + f];
  }
  for (int idx = tid; idx < NATOM * 32; idx += BLK) {
    int a = idx >> 5, f = idx & 31;
    sY[a * AS + f] = sX[a * AS + f];              // y0 = x[:,0,:] + segsum
  }
  __syncthreads();
  edge_fwd<true>(EPM, dstI, srcI, edgeBase, atomBase, sPos, sX, sY, sWr, sWb);
  __syncthreads();
  mm32(sY, AS, sT, AS, sW1, false, 64, sB1, 1);           // t1 = y0@w1 + b1
  __syncthreads();
  for (int idx = tid; idx < NATOM * 32; idx += BLK) {     // self-gated: t2 = t1*sig(t1)
    int a = idx >> 5, f = idx & 31;
    float t1v = sT[a * AS + f];
    t12w[(size_t)(atomBase + a) * 32 + f] = t1v;
    sY[a * AS + f] = t1v * sigm(t1v);
  }
  __syncthreads();
  mm32(sY, AS, sT, AS, sW2, false, 64, sB2, 1);           // t3 = t2@w2 + b2
  __syncthreads();

  // per-molecule energy: ae = (x0 + t3)@w_out + element_bias
  for (int a = wave * 8; a < wave * 8 + 8; ++a) {
    float xf = sX[a * AS + lane] + sT[a * AS + lane];
    float t = wave_sum(xf * sWout[lane]);
    if (lane == 0) atomicAdd(&sEnergy, t + ebias[Z[atomBase + a]]);
  }
  __syncthreads();
  if (tid == 0) out[mol] = sEnergy;

  // make ws activations visible across waves before the backward pass
  __threadfence();
  __syncthreads();
  __builtin_amdgcn_fence(__ATOMIC_ACQUIRE, "agent");

  // =========================== backward, iteration 2 =========================
  if (wave == 0) {                                // g_t2 = w_out @ w2^T (uniform)
    float acc = 0.f;
    for (int c = 0; c < 32; ++c) acc += sW2[lane * FS + c] * sWout[c];
    gt2v[lane] = acc;
  }
  for (int idx = tid; idx < NATOM * 32; idx += BLK) {
    int a = idx >> 5, f = idx & 31;
    sT[a * AS + f] = t12w[(size_t)(atomBase + a) * 32 + f];
  }
  __syncthreads();
  for (int idx = tid; idx < NATOM * 32; idx += BLK) {     // g_t1 (self-gate bwd)
    int a = idx >> 5, f = idx & 31;
    float t1v = sT[a * AS + f];
    float g = sigm(t1v);
    sY[a * AS + f] = gt2v[f] * (g + t1v * g * (1.f - g));
  }
  __syncthreads();
  mm32(sY, AS, sT, AS, sW1, true, 64, nullptr, 0);        // g_y0 = g_t1 @ w1^T
  __syncthreads();
  for (int idx = tid; idx < NATOM * AS; idx += BLK) sX[idx] = 0.f;  // gx buffer
  __syncthreads();
  for (int idx = tid; idx < NATOM * 32; idx += BLK) {
    int a = idx >> 5, f = idx & 31;
    sX[a * AS + f] = sWout[f] + sT[a * AS + f];           // g_x2[:,0,:] node paths
  }
  __syncthreads();
  edge_bwd<true>(EPM, dstI, srcI, edgeBase, atomBase, sPos, sT, sX, sGp,
                 xws + (size_t)2 * N * 128, sWr, sWb);
  __syncthreads();

  // ======================== backward, iterations 1, 0 ========================
  for (int it = 1; it >= 0; --it) {
    loadW(it, wr, wb, w1, b1, w2, b2, sWr, sWb, sW1, sW2, sB1, sB2);
    __syncthreads();
    mm32(sX, FS, sY, FS, sW2, true, 256, nullptr, 0);     // g_t2 = G @ w2^T
    for (int idx = tid; idx < NATOM * 128; idx += BLK) {  // t1_i from L2 scratch
      int a = idx >> 7, rem = idx & 127, l = rem >> 5, f = rem & 31;
      sT[a * AS + l * FS + f] =
          t1ws[(size_t)it * N * 128 + (size_t)(atomBase + a) * 128 + l * 32 + f];
    }
    __syncthreads();
    for (int a = wave * 8; a < wave * 8 + 8; ++a) {       // gate backward
      float t10 = sT[a * AS + lane];
      float g = sigm(t10);
      float q0 = sY[a * AS + lane],          q1 = sY[a * AS + FS + lane];
      float q2 = sY[a * AS + 2 * FS + lane], q3 = sY[a * AS + 3 * FS + lane];
      float u1 = sT[a * AS + FS + lane], u2 = sT[a * AS + 2 * FS + lane];
      float u3 = sT[a * AS + 3 * FS + lane];
      float gg = q0 * t10 + q1 * u1 + q2 * u2 + q3 * u3;  // grad into gate
      sY[a * AS + lane]          = q0 * g + gg * g * (1.f - g);
      sY[a * AS + FS + lane]     = q1 * g;
      sY[a * AS + 2 * FS + lane] = q2 * g;
      sY[a * AS + 3 * FS + lane] = q3 * g;
    }
    __syncthreads();
    mm32(sY, FS, sT, FS, sW1, true, 256, nullptr, 0);     // g_y = g_t1 @ w1^T
    __syncthreads();
    for (int idx = tid; idx < NATOM * 128; idx += BLK) {  // gx_i = G + g_y
      int a = idx >> 7, rem = idx & 127, l = rem >> 5, f = rem & 31;
      sX[a * AS + l * FS + f] += sT[a * AS + l * FS + f];
    }
    __syncthreads();
    edge_bwd<false>(EPM, dstI, srcI, edgeBase, atomBase, sPos, sT, sX, sGp,
                    xws + (size_t)it * N * 128, sWr, sWb);
    __syncthreads();
  }

  // forces = -d(sum E)/d pos
  for (int idx = tid; idx < NATOM * 3; idx += BLK)
    out[BS + (size_t)atomBase * 3 + idx] = -sGp[idx];
}

extern "C" void kernel_launch(void* const* d_in, const int* in_sizes, int n_in,
                              void* d_out, int out_size, void* d_ws, size_t ws_size,
                              hipStream_t stream) {
  const int*   Z     = (const int*)d_in[0];
  const float* pos   = (const float*)d_in[1];
  const int*   dstI  = (const int*)d_in[2];
  const int*   srcI  = (const int*)d_in[3];
  // d_in[4] batch_segments / d_in[5] batch_size: implied by block-contiguous layout
  const float* embed = (const float*)d_in[6];
  const float* wr    = (const float*)d_in[7];
  const float* wb    = (const float*)d_in[8];
  const float* w1    = (const float*)d_in[9];
  const float* b1    = (const float*)d_in[10];
  const float* w2    = (const float*)d_in[11];
  const float* b2    = (const float*)d_in[12];
  const float* wout  = (const float*)d_in[13];
  const float* ebias = (const float*)d_in[14];

  const int N   = in_sizes[0];          // 4096 atoms
  const int BS  = out_size - 3 * N;     // 64 molecules (out = energy + forces)
  const int NA  = N / BS;               // 64 atoms/molecule (kernel assumes 64)
  const int EPM = in_sizes[2] / BS;     // 4032 edges/molecule
  // ws usage: (3+2)*N*128 + N*32 floats = ~11 MB, fully L2-resident on MI455X.

  fisher_mpnn_kernel<<<BS, BLK, 0, stream>>>(Z, pos, dstI, srcI, embed, wr, wb,
      w1, b1, w2, b2, wout, ebias, (float*)d_out, (float*)d_ws, N, NA, EPM, BS);
}